// MiniOnsetRNN_29892972380793
// MI455X (gfx1250) — compile-verified
//
#include <hip/hip_runtime.h>

typedef __attribute__((ext_vector_type(2))) float v2f;
typedef __attribute__((ext_vector_type(8))) float v8f;

#define Bsz  64
#define Sseq 4096
#define Cin  80
#define Hh   10

// ---------------------------------------------------------------------------
// Kernel 1: x_proj[s][b][h] = sum_c input[b][s][c] * W_ih[h][c] + (b_ih+b_hh)
// GEMM view: A = input flattened to (B*S) x 80, B = W_ih^T (80 x 10, padded
// to 16 cols with zeros), done with V_WMMA_F32_16X16X4_F32, K = 80 = 20 x 4.
// One 16-row tile per wave, 4 waves per block. Memory-bound (84 MB stream).
// ---------------------------------------------------------------------------
__global__ __launch_bounds__(128) void xproj_wmma(const float* __restrict__ inp,
                                                  const float* __restrict__ Wih,
                                                  const float* __restrict__ bih,
                                                  const float* __restrict__ bhh,
                                                  float* __restrict__ xp) {
    const int lane = threadIdx.x & 31;
    const int wave = threadIdx.x >> 5;
    const int tile = blockIdx.x * 4 + wave;   // 16 rows of A per tile
    const int half = lane >> 4;               // lanes 0-15 vs 16-31
    const int n    = lane & 15;               // output column (h) for B/C/D
    const int rowA = tile * 16 + n;           // A row held by this lane
    const int nn   = (n < Hh) ? n : (Hh - 1); // clamped (avoid divergent load)
    const float nmask = (n < Hh) ? 1.0f : 0.0f;

    v8f acc = {};
    #pragma unroll
    for (int ks = 0; ks < Cin / 4; ++ks) {
        const int kk = ks * 4 + half * 2;
        // A fragment: lane holds A[rowA][kk], A[rowA][kk+1]
        const float2 a2 = *reinterpret_cast<const float2*>(inp + (size_t)rowA * Cin + kk);
        v2f afrag; afrag[0] = a2.x; afrag[1] = a2.y;
        // B fragment: B[k][n] = W_ih[n][k]; zero-pad columns n >= 10
        v2f bfrag;
        bfrag[0] = nmask * Wih[nn * Cin + kk];
        bfrag[1] = nmask * Wih[nn * Cin + kk + 1];
        acc = __builtin_amdgcn_wmma_f32_16x16x4_f32(
            false, afrag, false, bfrag, (short)0, acc, false, false);
    }

    if (n < Hh) {
        const float bias = bih[n] + bhh[n];
        #pragma unroll
        for (int v = 0; v < 8; ++v) {
            const int g    = tile * 16 + v + half * 8;  // flattened row = b*S + s
            const int bidx = g >> 12;                   // / 4096
            const int sidx = g & 4095;                  // % 4096
            xp[((size_t)sidx * Bsz + bidx) * Hh + n] = acc[v] + bias;
        }
    }
}

// ---------------------------------------------------------------------------
// Kernel 2: serial tanh recurrence + fused FC + sigmoid^4.
// Wave32-native mapping: lane = (half<<4) | i, two batches per wave
// (half 0: lanes 0-15, half 1: lanes 16-31), lane holds h_i of its batch.
// Per step: 10 x ds_bpermute (gather h_{t-1} of own half), 10 lane-local
// FMAs, 1 tanh. The FC output for step t-1 is computed from the SAME
// gathered array (every lane already holds all of h_{t-1}), so sigmoid^4 +
// store run in the shadow of the tanh dependency chain. 32 single-wave
// workgroups spread over 32 WGPs. xp loads pipelined 1 step + prefetch 16.
// ---------------------------------------------------------------------------
__device__ __forceinline__ float fast_tanh(float x) {
    const float e = __expf(2.0f * x);           // v_exp_f32 path
    return 1.0f - 2.0f / (e + 1.0f);
}

__device__ __forceinline__ float sig4(float logit) {
    const float sg = 1.0f / (1.0f + __expf(-logit));
    const float s2 = sg * sg;
    return s2 * s2;
}

__global__ __launch_bounds__(32) void rnn_scan(const float* __restrict__ xp,
                                               const float* __restrict__ Whh,
                                               const float* __restrict__ Wfc,
                                               const float* __restrict__ bfc_p,
                                               float* __restrict__ out) {
    const int lane = threadIdx.x;          // 0..31, one wave per block
    const int half = lane >> 4;            // which batch of this wave
    const int i    = lane & 15;            // h element owned by this lane
    const int ii   = (i < Hh) ? i : (Hh - 1);
    const int b    = blockIdx.x * 2 + half;          // batch element
    const int base = (lane & 16);                    // shuffle base lane

    // Row i of W_hh (lane-local), full W_fc (needed for redundant FC).
    float w[Hh], wf[Hh];
    #pragma unroll
    for (int j = 0; j < Hh; ++j) w[j]  = Whh[ii * Hh + j];
    #pragma unroll
    for (int j = 0; j < Hh; ++j) wf[j] = Wfc[j];
    const float bfc = bfc_p[0];

    float h = 0.0f;                        // h_{t-1}[i] for this batch

    // Prime pipeline: own xp element for t=0.
    float xc = xp[(size_t)b * Hh + ii];

    float* const outb = out + (size_t)b * Sseq;

    for (int t = 0; t < Sseq; ++t) {
        // Prefetch / pipelined load of xp for t+1.
        float xn = 0.0f;
        if (t + 1 < Sseq)
            xn = xp[((size_t)(t + 1) * Bsz + b) * Hh + ii];
        if (t + 16 < Sseq)
            __builtin_prefetch(xp + ((size_t)(t + 16) * Bsz + b) * Hh + ii, 0, 3);

        // Gather all of h_{t-1} for this half (10 x ds_bpermute).
        float hj[Hh];
        #pragma unroll
        for (int j = 0; j < Hh; ++j) hj[j] = __shfl(h, base + j, 32);

        // Critical chain: matvec (two 5-deep FMA chains) + tanh.
        float a0 = xc, a1 = 0.0f;
        #pragma unroll
        for (int j = 0; j < 5; ++j) a0 = fmaf(w[j],     hj[j],     a0);
        #pragma unroll
        for (int j = 0; j < 5; ++j) a1 = fmaf(w[j + 5], hj[j + 5], a1);
        const float nh = fast_tanh(a0 + a1);

        // Off-chain: FC + sigmoid^4 for timestep t-1 from gathered h_{t-1}.
        if (t > 0) {
            float l0 = bfc, l1 = 0.0f;
            #pragma unroll
            for (int j = 0; j < 5; ++j) l0 = fmaf(wf[j],     hj[j],     l0);
            #pragma unroll
            for (int j = 0; j < 5; ++j) l1 = fmaf(wf[j + 5], hj[j + 5], l1);
            if (i == 0) outb[t - 1] = sig4(l0 + l1);
        }

        h = nh;
        xc = xn;
    }

    // Epilogue: out[S-1] from the final hidden state.
    float hj[Hh];
    #pragma unroll
    for (int j = 0; j < Hh; ++j) hj[j] = __shfl(h, base + j, 32);
    float l0 = bfc, l1 = 0.0f;
    #pragma unroll
    for (int j = 0; j < 5; ++j) l0 = fmaf(wf[j],     hj[j],     l0);
    #pragma unroll
    for (int j = 0; j < 5; ++j) l1 = fmaf(wf[j + 5], hj[j + 5], l1);
    if (i == 0) outb[Sseq - 1] = sig4(l0 + l1);
}

// ---------------------------------------------------------------------------
extern "C" void kernel_launch(void* const* d_in, const int* in_sizes, int n_in,
                              void* d_out, int out_size, void* d_ws, size_t ws_size,
                              hipStream_t stream) {
    (void)in_sizes; (void)n_in; (void)out_size; (void)ws_size;
    const float* inp = (const float*)d_in[0];  // [64, 4096, 80]
    const float* Wih = (const float*)d_in[1];  // [10, 80]
    const float* Whh = (const float*)d_in[2];  // [10, 10]
    const float* bih = (const float*)d_in[3];  // [10]
    const float* bhh = (const float*)d_in[4];  // [10]
    const float* Wfc = (const float*)d_in[5];  // [1, 10]
    const float* bfc = (const float*)d_in[6];  // [1]
    float* out = (float*)d_out;                // [64, 4096, 1]
    float* xp  = (float*)d_ws;                 // [4096][64][10] = 10.49 MB

    const int tiles = (Bsz * Sseq) / 16;       // 16384 16-row tiles
    xproj_wmma<<<dim3(tiles / 4), 128, 0, stream>>>(inp, Wih, bih, bhh, xp);
    rnn_scan<<<dim3(Bsz / 2), 32, 0, stream>>>(xp, Whh, Wfc, bfc, out);
}